// QuanvolutionHybrid_65481071409421
// MI455X (gfx1250) — compile-verified
//
#include <hip/hip_runtime.h>
#include <math.h>

typedef __attribute__((ext_vector_type(2))) float v2f;
typedef __attribute__((ext_vector_type(8))) float v8f;

#define TILE_M   16
#define NPATCH   196
#define KDIM     784          // 196 patches * 4 feats
#define FSTRIDE  785          // odd stride -> conflict-free LDS reads of A
#define NCLASS   10
#define LSTRIDE  17

__global__ __launch_bounds__(128)
void quanv_fused_kernel(const float* __restrict__ x,
                        const float* __restrict__ params,
                        const float* __restrict__ W,
                        const float* __restrict__ bias,
                        float* __restrict__ out)
{
    __shared__ float featsL[TILE_M * FSTRIDE];   // 16 rows x 784 feats (padded)
    __shared__ float wt[KDIM * 16];              // W^T zero-padded to K x 16
    __shared__ float lgts[TILE_M * LSTRIDE];     // 16 x 16 logits tile (padded)

    const int tid = threadIdx.x;
    const int b0  = blockIdx.x * TILE_M;

    // Per-qubit constants: m_q(d) = cb[q]*cos(2d) - ka[q]*sin(2d)
    float cb[4], ka[4];
    #pragma unroll
    for (int q = 0; q < 4; ++q) {
        float alpha = params[2 * q];
        float beta  = params[2 * q + 1];
        cb[q] = cosf(beta);
        ka[q] = cosf(alpha) * sinf(beta);
    }

    // Stage W transposed into LDS as wt[k*16 + n], zero for n >= 10.
    for (int idx = tid; idx < KDIM * 16; idx += blockDim.x) {
        int n = idx & 15;
        int k = idx >> 4;
        wt[idx] = (n < NCLASS) ? W[n * KDIM + k] : 0.0f;
    }

    // Compute features for 16 batch rows into LDS.
    for (int t = tid; t < TILE_M * NPATCH; t += blockDim.x) {
        int m  = t / NPATCH;
        int p  = t - m * NPATCH;
        int r2 = p / 14;
        int c2 = p - r2 * 14;
        const float* xb = x + (size_t)(b0 + m) * KDIM + (r2 * 56 + c2 * 2);
        float d0 = xb[0], d1 = xb[1], d2 = xb[28], d3 = xb[29];

        float s, c;
        sincosf(2.0f * d0, &s, &c); float m0 = cb[0] * c - ka[0] * s;
        sincosf(2.0f * d1, &s, &c); float m1 = cb[1] * c - ka[1] * s;
        sincosf(2.0f * d2, &s, &c); float m2 = cb[2] * c - ka[2] * s;
        sincosf(2.0f * d3, &s, &c); float m3 = cb[3] * c - ka[3] * s;

        float* f = featsL + m * FSTRIDE + p * 4;
        f[0] = m0;
        f[1] = m0 * m1;
        f[2] = m2;
        f[3] = m2 * m3;
    }
    __syncthreads();

    // Wave 0: 16x16 logits tile via V_WMMA_F32_16X16X4_F32 over K=784.
    if (tid < 32) {
        const int nm   = tid & 15;   // M for A, N for B/C/D
        const int half = tid >> 4;   // lane group: selects K+0/1 vs K+2/3
        const float* fa = featsL + nm * FSTRIDE + 2 * half;
        const float* fb = wt + (2 * half) * 16 + nm;

        v8f acc = {};
        #pragma unroll 4
        for (int k0 = 0; k0 < KDIM; k0 += 4) {
            v2f A;  A.x  = fa[k0];        A.y  = fa[k0 + 1];
            v2f Bv; Bv.x = fb[k0 * 16];   Bv.y = fb[k0 * 16 + 16];
            acc = __builtin_amdgcn_wmma_f32_16x16x4_f32(
                /*neg_a=*/false, A, /*neg_b=*/false, Bv,
                /*c_mod=*/(short)0, acc, /*reuse_a=*/false, /*reuse_b=*/false);
        }
        // D layout: VGPR r holds row (r + 8*half), col = lane&15
        #pragma unroll
        for (int r = 0; r < 8; ++r)
            lgts[(r + 8 * half) * LSTRIDE + nm] = acc[r];
    }
    __syncthreads();

    // Per-row log_softmax over the 10 valid classes.
    if (tid < TILE_M) {
        const int row = tid;
        float lg[NCLASS];
        float mx = -3.0e38f;
        #pragma unroll
        for (int j = 0; j < NCLASS; ++j) {
            lg[j] = lgts[row * LSTRIDE + j] + bias[j];
            mx = fmaxf(mx, lg[j]);
        }
        float sum = 0.0f;
        #pragma unroll
        for (int j = 0; j < NCLASS; ++j) sum += expf(lg[j] - mx);
        float lse = logf(sum);
        float* o = out + (size_t)(b0 + row) * NCLASS;
        #pragma unroll
        for (int j = 0; j < NCLASS; ++j) o[j] = lg[j] - mx - lse;
    }
}

extern "C" void kernel_launch(void* const* d_in, const int* in_sizes, int n_in,
                              void* d_out, int out_size, void* d_ws, size_t ws_size,
                              hipStream_t stream) {
    (void)n_in; (void)out_size; (void)d_ws; (void)ws_size;
    const float* x      = (const float*)d_in[0];   // (4096,1,28,28)
    const float* params = (const float*)d_in[1];   // (8,)
    const float* W      = (const float*)d_in[2];   // (10,784)
    const float* bias   = (const float*)d_in[3];   // (10,)
    float* out          = (float*)d_out;           // (4096,10)

    const int B      = in_sizes[0] / KDIM;         // 4096
    const int blocks = B / TILE_M;                 // 256

    quanv_fused_kernel<<<blocks, 128, 0, stream>>>(x, params, W, bias, out);
}